// OutputModule_44813688767175
// MI455X (gfx1250) — compile-verified
//
#include <hip/hip_runtime.h>
#include <hip/hip_bf16.h>

// Problem sizes (fixed by the reference)
#define G_    256
#define N_    2048
#define C_    512
#define POOL_ 32
#define HID_  64

typedef __attribute__((ext_vector_type(2))) float v2f;
typedef __attribute__((ext_vector_type(4))) float v4f;
typedef __attribute__((ext_vector_type(8))) float v8f;

// ---------------------------------------------------------------------------
// Kernel 1: adaptive average pool.
// x: [G*N, C] f32 row-major. pooled[g,p,c] = mean over 64 rows.
// One block per (g,p): since g*2048 + p*64 == blockIdx.x*64, addressing is flat.
// 256 threads: thread t owns float4-column (t&127), row-half (t>>7).
// Non-temporal loads: x is 1 GB streamed once; keep L2 clean for weights.
// This kernel IS the roofline: ~1.07 GB / 23.3 TB/s ~= 46 us.
// ---------------------------------------------------------------------------
__global__ __launch_bounds__(256)
void pool_kernel(const float* __restrict__ x, float* __restrict__ pooled)
{
    const int t    = threadIdx.x;
    const int col  = t & 127;     // float4 column, 128 * 4 = 512 channels
    const int half = t >> 7;      // which 32-row half
    const v4f* px = (const v4f*)x
                  + (size_t)blockIdx.x * (64 * 128)
                  + (size_t)half * (32 * 128)
                  + col;
    v4f acc = {};
    #pragma unroll 4
    for (int r = 0; r < 32; ++r) {
        v4f a = __builtin_nontemporal_load(px);
        px += 128;
        acc += a;
    }
    __shared__ v4f sh[256];
    sh[t] = acc;
    __syncthreads();
    if (t < 128) {
        v4f o = (sh[t] + sh[t + 128]) * (1.0f / 64.0f);
        ((v4f*)pooled)[(size_t)blockIdx.x * 128 + t] = o;   // pooled[g][p][c]
    }
}

// ---------------------------------------------------------------------------
// Kernels 2/3: Conv1d(k=3, pad=SAME) + bias + ReLU as im2col GEMM on
// V_WMMA_F32_16X16X4_F32.  out[m,n] = sum_k W[m,k] * P[n + (k%3) - 1, k/3]
// with A[m,k] == W flat ([Cout, Cin, 3] row-major).
//
// Input tile staged in LDS with zero-padded boundary rows (rows 0 and L+1)
// and bank-conflict-free row stride PS (PS % 64 == 4 -> 4-bank stride, still
// 16B aligned for B128 async staging). One block per graph; each wave owns
// one 16x16 output tile.
//
// K-loop: the im2col (d = k%3, c = k/3) pattern repeats with period 3 in the
// WMMA step index (K advances 12 -> c advances 4, d unchanged), so we
// precompute 6 per-lane LDS base pointers and the loop is purely affine:
// 6x ds_load_b32 + 3x global b64 (A) + 3x v_wmma per group.
//
//   conv1: CIN=512 COUT=64 -> 8 waves (BLOCK=256), 128 groups (384 WMMA)
//   conv2: CIN=64  COUT=32 -> 4 waves (BLOCK=128), 16 groups  (48 WMMA)
//
// POS_MAJOR (conv1): input [pos][chan], contiguous rows -> staged with
// GLOBAL_LOAD_ASYNC_TO_LDS_B128 (memory->LDS DMA path, ASYNCcnt).
// conv2 input is [chan][pos] and must transpose-scatter -> plain staging.
// ---------------------------------------------------------------------------
template<int CIN, int COUT, bool POS_MAJOR, int PS, int BLOCK>
__global__ __launch_bounds__(BLOCK)
void conv1d_relu_wmma(const float* __restrict__ in,
                      const float* __restrict__ W,     // [COUT][CIN*3] flat
                      const float* __restrict__ bias,  // [COUT]
                      float* __restrict__ out)         // [G][COUT][32]
{
    constexpr int L = 32;
    constexpr int K = CIN * 3;
    static_assert(K % 12 == 0, "K-loop grouped by 3 WMMA steps");
    extern __shared__ float ldsP[];                    // (L+2)*PS floats

    const int g = blockIdx.x;
    const int t = threadIdx.x;

    // zero-pad boundary rows (SAME padding)
    for (int i = t; i < PS; i += BLOCK) {
        ldsP[i] = 0.0f;
        ldsP[(L + 1) * PS + i] = 0.0f;
    }

    // stage input: ldsP[(pos+1)*PS + c]
    const float* ing = in + (size_t)g * (L * CIN);
    if constexpr (POS_MAJOR) {
        // async memory->LDS B128 staging (row-contiguous both sides)
        const uint32_t lds0 = (uint32_t)(uintptr_t)ldsP;  // low 32 bits of the
        // flat shared pointer == wave-relative LDS byte offset
        constexpr int NCH4 = CIN / 4;                     // float4s per row
        for (int i4 = t; i4 < L * NCH4; i4 += BLOCK) {
            const int pos = i4 / NCH4;
            const int c4  = i4 - pos * NCH4;
            const uint32_t laddr = lds0 + (uint32_t)(((pos + 1) * PS + c4 * 4) * 4);
            const uint64_t gaddr = (uint64_t)(uintptr_t)(ing + (size_t)i4 * 4);
            asm volatile("global_load_async_to_lds_b128 %0, %1, off"
                         :: "v"(laddr), "v"(gaddr) : "memory");
        }
        asm volatile("s_wait_asynccnt 0x0" ::: "memory");
    } else {
        // transpose-scatter: in[c][pos] -> ldsP[(pos+1)*PS + c]
        for (int i = t; i < L * CIN; i += BLOCK) {
            const int c   = i / L;
            const int pos = i - c * L;
            ldsP[(pos + 1) * PS + c] = ing[i];
        }
    }
    __syncthreads();

    const int wave  = t >> 5;       // wave32
    const int lane  = t & 31;
    const int half  = lane >> 4;    // K sub-pair selector per WMMA layout
    const int lh    = lane & 15;
    const int mtile = wave >> 1;    // 16-row tile of Cout
    const int ntile = wave & 1;     // 16-col tile of L
    const int m = mtile * 16 + lh;  // A row for this lane
    const int n = ntile * 16 + lh;  // B/D column for this lane

    // A[m, 12*kkk + 4*p + 2*half + {0,1}] as an 8B-aligned float2 stream
    const v2f* ap = (const v2f*)(W + (size_t)m * K);

    // 6 periodic B base pointers: slot s = 2*p + which, koff = k mod 12
    const float* bb[6];
    #pragma unroll
    for (int s = 0; s < 6; ++s) {
        const int koff = 4 * (s >> 1) + (s & 1) + 2 * half;  // [0, 11]
        const int d  = koff % 3;
        const int cf = koff / 3;
        bb[s] = ldsP + (n + d) * PS + cf;   // row (n+d-1) via +1 padded offset
    }

    v8f acc = {};
    #pragma unroll 2
    for (int kkk = 0; kkk < K / 12; ++kkk) {
        #pragma unroll
        for (int p = 0; p < 3; ++p) {
            v2f a = ap[6 * kkk + 2 * p + half];
            v2f b;
            b.x = bb[2 * p + 0][4 * kkk];
            b.y = bb[2 * p + 1][4 * kkk];
            acc = __builtin_amdgcn_wmma_f32_16x16x4_f32(
                      false, a, false, b, (short)0, acc, false, false);
        }
    }

    // D layout: VGPR j -> row mtile*16 + half*8 + j, col n
    float* og = out + (size_t)g * (COUT * L);
    #pragma unroll
    for (int j = 0; j < 8; ++j) {
        const int mm = mtile * 16 + half * 8 + j;
        float v = acc[j] + bias[mm];
        og[mm * L + n] = fmaxf(v, 0.0f);
    }
}

// ---------------------------------------------------------------------------
// Kernel 4: fc1 -> LN -> ReLU -> fc2 -> LN -> ReLU -> out-projection.
// One 64-thread block per graph; thread j owns hidden unit j.  ~0.04 GFLOP
// total across 256 graphs: wall-clock irrelevant, keep it simple and exact.
// ---------------------------------------------------------------------------
__global__ __launch_bounds__(64)
void head_kernel(const float* __restrict__ h2,     // [G][1024]
                 const float* __restrict__ fc1_w, const float* __restrict__ fc1_b,
                 const float* __restrict__ fc2_w, const float* __restrict__ fc2_b,
                 const float* __restrict__ ln_g,  const float* __restrict__ ln_b,
                 const float* __restrict__ out_w, const float* __restrict__ out_b,
                 float* __restrict__ outp)         // [G]
{
    __shared__ float sv[1024];
    __shared__ float sh[64];
    const int g = blockIdx.x, j = threadIdx.x;

    const float* v = h2 + (size_t)g * 1024;
    for (int i = j; i < 1024; i += 64) sv[i] = v[i];
    __syncthreads();

    // fc1
    float acc = fc1_b[j];
    const float* wr = fc1_w + (size_t)j * 1024;
    #pragma unroll 8
    for (int k = 0; k < 1024; ++k) acc = fmaf(wr[k], sv[k], acc);
    sh[j] = acc;
    __syncthreads();
    // LayerNorm(64) + ReLU
    float mu = 0.0f;
    for (int k = 0; k < 64; ++k) mu += sh[k];
    mu *= (1.0f / 64.0f);
    float var = 0.0f;
    for (int k = 0; k < 64; ++k) { float d = sh[k] - mu; var = fmaf(d, d, var); }
    var *= (1.0f / 64.0f);
    float inv = rsqrtf(var + 1e-5f);
    float h = fmaxf((acc - mu) * inv * ln_g[j] + ln_b[j], 0.0f);
    __syncthreads();
    sh[j] = h;
    __syncthreads();

    // fc2
    float acc2 = fc2_b[j];
    const float* w2 = fc2_w + (size_t)j * 64;
    #pragma unroll 8
    for (int k = 0; k < 64; ++k) acc2 = fmaf(w2[k], sh[k], acc2);
    __syncthreads();
    sh[j] = acc2;
    __syncthreads();
    mu = 0.0f;
    for (int k = 0; k < 64; ++k) mu += sh[k];
    mu *= (1.0f / 64.0f);
    var = 0.0f;
    for (int k = 0; k < 64; ++k) { float d = sh[k] - mu; var = fmaf(d, d, var); }
    var *= (1.0f / 64.0f);
    inv = rsqrtf(var + 1e-5f);
    h = fmaxf((acc2 - mu) * inv * ln_g[j] + ln_b[j], 0.0f);
    __syncthreads();
    sh[j] = h;
    __syncthreads();

    if (j == 0) {
        float o = out_b[0];
        for (int k = 0; k < 64; ++k) o = fmaf(out_w[k], sh[k], o);
        outp[g] = o;
    }
}

// ---------------------------------------------------------------------------
extern "C" void kernel_launch(void* const* d_in, const int* in_sizes, int n_in,
                              void* d_out, int out_size, void* d_ws, size_t ws_size,
                              hipStream_t stream)
{
    (void)in_sizes; (void)n_in; (void)out_size; (void)ws_size;

    const float* x       = (const float*)d_in[0];
    // d_in[1] = batch (int64): layout is fixed (sorted, equal-sized) -> unused
    const float* conv1_w = (const float*)d_in[2];
    const float* conv1_b = (const float*)d_in[3];
    const float* conv2_w = (const float*)d_in[4];
    const float* conv2_b = (const float*)d_in[5];
    const float* fc1_w   = (const float*)d_in[6];
    const float* fc1_b   = (const float*)d_in[7];
    const float* fc2_w   = (const float*)d_in[8];
    const float* fc2_b   = (const float*)d_in[9];
    const float* ln_g    = (const float*)d_in[10];
    const float* ln_b    = (const float*)d_in[11];
    const float* out_w   = (const float*)d_in[12];
    const float* out_b   = (const float*)d_in[13];

    // Workspace: pooled [G,POOL,C] (16 MB) | h1 [G,64,32] (2 MB) | h2 [G,32,32] (1 MB)
    float* pooled = (float*)d_ws;
    float* h1 = pooled + (size_t)G_ * POOL_ * C_;
    float* h2 = h1     + (size_t)G_ * HID_  * POOL_;

    pool_kernel<<<G_ * POOL_, 256, 0, stream>>>(x, pooled);

    // conv1: CIN=512, COUT=64, pos-major input, PS=516 (34*516*4 = 70176 B LDS)
    conv1d_relu_wmma<512, 64, true, 516, 256>
        <<<G_, 256, 34 * 516 * sizeof(float), stream>>>(pooled, conv1_w, conv1_b, h1);

    // conv2: CIN=64, COUT=32, chan-major input, PS=68 (34*68*4 = 9248 B LDS)
    conv1d_relu_wmma<64, 32, false, 68, 128>
        <<<G_, 128, 34 * 68 * sizeof(float), stream>>>(h1, conv2_w, conv2_b, h2);

    head_kernel<<<G_, 64, 0, stream>>>(h2, fc1_w, fc1_b, fc2_w, fc2_b,
                                       ln_g, ln_b, out_w, out_b, (float*)d_out);
}